// AttentionBlock_38568806318848
// MI455X (gfx1250) — compile-verified
//
#include <hip/hip_runtime.h>
#include <math.h>

typedef __attribute__((ext_vector_type(2))) float v2f;
typedef __attribute__((ext_vector_type(8))) float v8f;

// ---------------------------------------------------------------------------
// Problem constants (from reference): B=4, C=512, H=W=64
// ---------------------------------------------------------------------------
#define BB     4
#define CIN    512
#define HW     4096           // 64*64
#define HWP    1024           // 32*32 pooled
#define CQK    64             // C/8
#define CH     256            // C/2
#define CCAT   384            // 64 + 64 + 256 stacked channels
#define COUT   512

__device__ inline v8f wmma4(v2f a, v2f b, v8f c) {
  // V_WMMA_F32_16X16X4_F32 : D = A(16x4,f32) * B(4x16,f32) + C(16x16,f32)
  return __builtin_amdgcn_wmma_f32_16x16x4_f32(
      false, a, false, b, (short)0, c, false, false);
}

__device__ inline v8f vzero8() {
  v8f z;
#pragma unroll
  for (int r = 0; r < 8; ++r) z[r] = 0.0f;
  return z;
}

// ---------------------------------------------------------------------------
// Kernel P: pack [wf;wg;wh] -> wcat[384][512], gf/gg/gh -> gcat, bf/bg/bh -> bcat
// ---------------------------------------------------------------------------
__global__ __launch_bounds__(256) void pack_w(
    const float* __restrict__ wf, const float* __restrict__ wg,
    const float* __restrict__ wh, const float* __restrict__ gf,
    const float* __restrict__ bf, const float* __restrict__ gg,
    const float* __restrict__ bg, const float* __restrict__ gh,
    const float* __restrict__ bh, float* __restrict__ wcat,
    float* __restrict__ gcat, float* __restrict__ bcat) {
  int i = blockIdx.x * 256 + threadIdx.x;
  if (i < CCAT * CIN) {
    int r = i >> 9, c = i & 511;
    float v;
    if (r < 64)       v = wf[r * CIN + c];
    else if (r < 128) v = wg[(r - 64) * CIN + c];
    else              v = wh[(r - 128) * CIN + c];
    wcat[i] = v;
  }
  if (i < CCAT) {
    float g, b;
    if (i < 64)       { g = gf[i];        b = bf[i]; }
    else if (i < 128) { g = gg[i - 64];   b = bg[i - 64]; }
    else              { g = gh[i - 128];  b = bh[i - 128]; }
    gcat[i] = g; bcat[i] = b;
  }
}

// ---------------------------------------------------------------------------
// Kernel A: y1[b][384][4096] = wcat[384][512] @ x[b][512][4096]   (raw, pre-BN)
// Workgroup tile 128(M) x 128(N); 8 waves as 4(M) x 2(N); wave = 32x64
// (2 A-frags amortized over 4 B-frags -> 0.75 LDS frags per WMMA).
// ---------------------------------------------------------------------------
__global__ __launch_bounds__(256) void gemm_qkv(
    const float* __restrict__ x, const float* __restrict__ wcat,
    float* __restrict__ y1) {
  __shared__ float Wt[128][33];
  __shared__ float Xt[32][129];
  const int b  = blockIdx.z;
  const int n0 = blockIdx.x * 128;
  const int m0 = blockIdx.y * 128;
  const int t = threadIdx.x, lane = t & 31, wave = t >> 5;
  const int wm = wave >> 1, wn = wave & 1;
  const int kq = (lane >> 4) << 1, l15 = lane & 15, hi8 = (lane >> 4) << 3;
  const float* xb = x + (size_t)b * CIN * HW;

  v8f acc[2][4];
#pragma unroll
  for (int i = 0; i < 2; ++i)
#pragma unroll
    for (int j = 0; j < 4; ++j) acc[i][j] = vzero8();

  for (int kb = 0; kb < CIN; kb += 32) {
    for (int e = t; e < 128 * 32; e += 256) {
      int mi = e >> 5, ki = e & 31;
      Wt[mi][ki] = wcat[(m0 + mi) * CIN + kb + ki];
    }
    for (int e = t; e < 32 * 128; e += 256) {
      int ki = e >> 7, nj = e & 127;
      Xt[ki][nj] = xb[(size_t)(kb + ki) * HW + n0 + nj];
    }
    __syncthreads();
#pragma unroll
    for (int kk = 0; kk < 32; kk += 4) {
      int k0 = kk + kq;
      v2f a[2];
#pragma unroll
      for (int i = 0; i < 2; ++i) {
        a[i].x = Wt[wm * 32 + i * 16 + l15][k0];
        a[i].y = Wt[wm * 32 + i * 16 + l15][k0 + 1];
      }
#pragma unroll
      for (int j = 0; j < 4; ++j) {
        int nn = wn * 64 + j * 16 + l15;
        v2f bv; bv.x = Xt[k0][nn]; bv.y = Xt[k0 + 1][nn];
#pragma unroll
        for (int i = 0; i < 2; ++i) acc[i][j] = wmma4(a[i], bv, acc[i][j]);
      }
    }
    __syncthreads();
  }
  float* yb = y1 + (size_t)b * CCAT * HW;
#pragma unroll
  for (int i = 0; i < 2; ++i) {
    int mrow0 = m0 + wm * 32 + i * 16 + hi8;
#pragma unroll
    for (int j = 0; j < 4; ++j) {
      int nc = n0 + wn * 64 + j * 16 + l15;
#pragma unroll
      for (int r = 0; r < 8; ++r)
        yb[(size_t)(mrow0 + r) * HW + nc] = acc[i][j][r];
    }
  }
}

// ---------------------------------------------------------------------------
// Kernel B: batch-norm stats per channel over (B, N) -> fused scale/shift
// ---------------------------------------------------------------------------
__global__ __launch_bounds__(256) void bn_stats(
    const float* __restrict__ y, const float* __restrict__ g,
    const float* __restrict__ b, float* __restrict__ scale,
    float* __restrict__ shift, int C) {
  const int ch = blockIdx.x, t = threadIdx.x;
  float s = 0.f, s2 = 0.f;
  for (int bb = 0; bb < BB; ++bb) {
    const float* p = y + ((size_t)bb * C + ch) * HW;
    for (int n = t; n < HW; n += 256) { float v = p[n]; s += v; s2 += v * v; }
  }
  __shared__ float sh[256], sh2[256];
  sh[t] = s; sh2[t] = s2; __syncthreads();
  for (int o = 128; o > 0; o >>= 1) {
    if (t < o) { sh[t] += sh[t + o]; sh2[t] += sh2[t + o]; }
    __syncthreads();
  }
  if (t == 0) {
    const float inv = 1.0f / (float)(BB * HW);
    float mean = sh[0] * inv;
    float var  = sh2[0] * inv - mean * mean;
    float sc   = g[ch] * rsqrtf(var + 1e-5f);
    scale[ch] = sc;
    shift[ch] = b[ch] - mean * sc;
  }
}

// ---------------------------------------------------------------------------
// Kernel C: BN+ReLU; maxpool f and h; write g_act[c][n], f_pool[c][m], ht[m][c]
// ---------------------------------------------------------------------------
__global__ __launch_bounds__(256) void act_pool(
    const float* __restrict__ y1, const float* __restrict__ scale,
    const float* __restrict__ shift, float* __restrict__ gact,
    float* __restrict__ fpool, float* __restrict__ ht) {
  const int PER_B = CQK * HW + CQK * HWP + CH * HWP;  // 589824
  int idx = blockIdx.x * 256 + threadIdx.x;
  if (idx >= BB * PER_B) return;
  int b = idx / PER_B, r = idx % PER_B;
  const float* yb = y1 + (size_t)b * CCAT * HW;
  if (r < CQK * HW) {                       // g branch (channels 64..127)
    int c = r >> 12, n = r & (HW - 1);
    int ch = 64 + c;
    float v = yb[(size_t)ch * HW + n] * scale[ch] + shift[ch];
    gact[((size_t)b * CQK + c) * HW + n] = fmaxf(v, 0.f);
  } else if (r < CQK * HW + CQK * HWP) {    // f branch pooled (channels 0..63)
    int rr = r - CQK * HW;
    int c = rr >> 10, m = rr & (HWP - 1);
    int h2 = m >> 5, w2 = m & 31;
    int n00 = (h2 * 2) * 64 + w2 * 2;
    float sc = scale[c], sf = shift[c];
    const float* p = yb + (size_t)c * HW;
    float v0 = fmaxf(p[n00] * sc + sf, 0.f);
    float v1 = fmaxf(p[n00 + 1] * sc + sf, 0.f);
    float v2 = fmaxf(p[n00 + 64] * sc + sf, 0.f);
    float v3 = fmaxf(p[n00 + 65] * sc + sf, 0.f);
    fpool[((size_t)b * CQK + c) * HWP + m] = fmaxf(fmaxf(v0, v1), fmaxf(v2, v3));
  } else {                                  // h branch pooled+transposed (128..383)
    int rr = r - CQK * HW - CQK * HWP;
    int c = rr >> 10, m = rr & (HWP - 1);
    int ch = 128 + c;
    int h2 = m >> 5, w2 = m & 31;
    int n00 = (h2 * 2) * 64 + w2 * 2;
    float sc = scale[ch], sf = shift[ch];
    const float* p = yb + (size_t)ch * HW;
    float v0 = fmaxf(p[n00] * sc + sf, 0.f);
    float v1 = fmaxf(p[n00 + 1] * sc + sf, 0.f);
    float v2 = fmaxf(p[n00 + 64] * sc + sf, 0.f);
    float v3 = fmaxf(p[n00 + 65] * sc + sf, 0.f);
    ht[((size_t)b * HWP + m) * CH + c] = fmaxf(fmaxf(v0, v1), fmaxf(v2, v3));
  }
}

// ---------------------------------------------------------------------------
// Kernel D: attention per batch. 64 query rows per workgroup.
// Pass 1: streaming S = G^T F tiles (WMMA K=64) -> exact row max / sumexp.
// Pass 2: recompute S, P=exp(s-max)/sum staged via LDS, O += P @ H^T (WMMA).
// ---------------------------------------------------------------------------
__global__ __launch_bounds__(256) void attention(
    const float* __restrict__ gact, const float* __restrict__ fpool,
    const float* __restrict__ ht, float* __restrict__ oatt) {
  __shared__ float Gt[64][65];     // G tile: [c][n_local], resident
  __shared__ float U[32 * 256];    // union: Ft[64][33] (2112) | Htile[32][256]
  __shared__ float Ps[64][33];     // staged P block [n_local][m_local]
  __shared__ float rmax[64], rrcp[64];
  __shared__ float cmax[2][64], csum[2][64];

  const int b  = blockIdx.y;
  const int nb = blockIdx.x * 64;
  const int t = threadIdx.x, lane = t & 31, wave = t >> 5;
  const int wn = wave >> 1, wh = wave & 1;  // wh: m-half (S) / c-half (O)
  const int kq = (lane >> 4) << 1, l15 = lane & 15, hi8 = (lane >> 4) << 3;

  const float* gb = gact + (size_t)b * CQK * HW;
  const float* fb = fpool + (size_t)b * CQK * HWP;
  const float* hb = ht + (size_t)b * HWP * CH;

  for (int e = t; e < 64 * 64; e += 256) {
    int c = e >> 6, ni = e & 63;
    Gt[c][ni] = gb[(size_t)c * HW + nb + ni];
  }
  __syncthreads();

  // ---------------- pass 1: online row max & sum of exp -------------------
  float mrun[8], lrun[8];
#pragma unroll
  for (int r = 0; r < 8; ++r) { mrun[r] = -3.0e38f; lrun[r] = 0.f; }

  for (int mb = 0; mb < HWP; mb += 32) {
    for (int e = t; e < 64 * 32; e += 256) {
      int mi = e & 31, c = e >> 5;
      U[c * 33 + mi] = fb[(size_t)c * HWP + mb + mi];
    }
    __syncthreads();
    v8f s = vzero8();
#pragma unroll
    for (int kk = 0; kk < 64; kk += 4) {
      int k0 = kk + kq;
      v2f a;  a.x  = Gt[k0][wn * 16 + l15];       a.y  = Gt[k0 + 1][wn * 16 + l15];
      v2f bv; bv.x = U[k0 * 33 + wh * 16 + l15];  bv.y = U[(k0 + 1) * 33 + wh * 16 + l15];
      s = wmma4(a, bv, s);
    }
#pragma unroll
    for (int r = 0; r < 8; ++r) {
      float v = s[r];
      float tm = v;
#pragma unroll
      for (int o = 1; o < 16; o <<= 1) tm = fmaxf(tm, __shfl_xor(tm, o, 32));
      float nm = fmaxf(mrun[r], tm);
      float ts = __expf(v - nm);
#pragma unroll
      for (int o = 1; o < 16; o <<= 1) ts += __shfl_xor(ts, o, 32);
      lrun[r] = lrun[r] * __expf(mrun[r] - nm) + ts;
      mrun[r] = nm;
    }
    __syncthreads();
  }
  if (l15 == 0) {
#pragma unroll
    for (int r = 0; r < 8; ++r) {
      cmax[wh][wn * 16 + r + hi8] = mrun[r];
      csum[wh][wn * 16 + r + hi8] = lrun[r];
    }
  }
  __syncthreads();
  if (t < 64) {
    float m0 = cmax[0][t], m1 = cmax[1][t];
    float M = fmaxf(m0, m1);
    float S = csum[0][t] * __expf(m0 - M) + csum[1][t] * __expf(m1 - M);
    rmax[t] = M; rrcp[t] = 1.0f / S;
  }
  __syncthreads();

  // ---------------- pass 2: O = softmax(S) @ H^T --------------------------
  v8f oacc[8];
#pragma unroll
  for (int j = 0; j < 8; ++j) oacc[j] = vzero8();

  for (int mb = 0; mb < HWP; mb += 32) {
    for (int e = t; e < 64 * 32; e += 256) {   // reload F tile into union
      int mi = e & 31, c = e >> 5;
      U[c * 33 + mi] = fb[(size_t)c * HWP + mb + mi];
    }
    __syncthreads();
    v8f s = vzero8();
#pragma unroll
    for (int kk = 0; kk < 64; kk += 4) {
      int k0 = kk + kq;
      v2f a;  a.x  = Gt[k0][wn * 16 + l15];       a.y  = Gt[k0 + 1][wn * 16 + l15];
      v2f bv; bv.x = U[k0 * 33 + wh * 16 + l15];  bv.y = U[(k0 + 1) * 33 + wh * 16 + l15];
      s = wmma4(a, bv, s);
    }
#pragma unroll
    for (int r = 0; r < 8; ++r) {              // stage P tile (D-layout -> LDS)
      int nl = wn * 16 + r + hi8;
      Ps[nl][wh * 16 + l15] = __expf(s[r] - rmax[nl]) * rrcp[nl];
    }
    __syncthreads();
    for (int e = t; e < 32 * 256; e += 256) {  // H^T tile overwrites union
      int c = e & 255, mi = e >> 8;
      U[mi * 256 + c] = hb[(size_t)(mb + mi) * CH + c];
    }
    __syncthreads();
#pragma unroll
    for (int kk = 0; kk < 32; kk += 4) {       // O += P @ H^T
      int k0 = kk + kq;
      v2f a; a.x = Ps[wn * 16 + l15][k0]; a.y = Ps[wn * 16 + l15][k0 + 1];
#pragma unroll
      for (int j = 0; j < 8; ++j) {
        int cc = wh * 128 + j * 16 + l15;
        v2f bv; bv.x = U[k0 * 256 + cc]; bv.y = U[(k0 + 1) * 256 + cc];
        oacc[j] = wmma4(a, bv, oacc[j]);
      }
    }
    __syncthreads();
  }
  float* ob = oatt + (size_t)b * HW * CH;
#pragma unroll
  for (int j = 0; j < 8; ++j) {
    int cc = wh * 128 + j * 16 + l15;
#pragma unroll
    for (int r = 0; r < 8; ++r) {
      int nl = wn * 16 + r + hi8;
      ob[(size_t)(nb + nl) * CH + cc] = oacc[j][r];
    }
  }
}

// ---------------------------------------------------------------------------
// Kernel E: y2[b][512][4096] = wo[512][256] @ oatt[b][.][.]^T  (raw, pre-BN)
// Workgroup tile 128(M) x 128(N); wave = 32x64. B-tile read from oatt[n][c]
// coalesced, transposed in LDS.
// ---------------------------------------------------------------------------
__global__ __launch_bounds__(256) void gemm_out(
    const float* __restrict__ oatt, const float* __restrict__ wo,
    float* __restrict__ y2) {
  __shared__ float Wt[128][33];
  __shared__ float Xt[32][129];
  const int b  = blockIdx.z;
  const int n0 = blockIdx.x * 128;
  const int m0 = blockIdx.y * 128;
  const int t = threadIdx.x, lane = t & 31, wave = t >> 5;
  const int wm = wave >> 1, wn = wave & 1;
  const int kq = (lane >> 4) << 1, l15 = lane & 15, hi8 = (lane >> 4) << 3;
  const float* ob = oatt + (size_t)b * HW * CH;

  v8f acc[2][4];
#pragma unroll
  for (int i = 0; i < 2; ++i)
#pragma unroll
    for (int j = 0; j < 4; ++j) acc[i][j] = vzero8();

  for (int kb = 0; kb < CH; kb += 32) {
    for (int e = t; e < 128 * 32; e += 256) {
      int mi = e >> 5, ki = e & 31;
      Wt[mi][ki] = wo[(m0 + mi) * CH + kb + ki];
    }
    for (int e = t; e < 32 * 128; e += 256) {
      int ci = e & 31, nj = e >> 5;                 // coalesced over ci
      Xt[ci][nj] = ob[(size_t)(n0 + nj) * CH + kb + ci];
    }
    __syncthreads();
#pragma unroll
    for (int kk = 0; kk < 32; kk += 4) {
      int k0 = kk + kq;
      v2f a[2];
#pragma unroll
      for (int i = 0; i < 2; ++i) {
        a[i].x = Wt[wm * 32 + i * 16 + l15][k0];
        a[i].y = Wt[wm * 32 + i * 16 + l15][k0 + 1];
      }
#pragma unroll
      for (int j = 0; j < 4; ++j) {
        int nn = wn * 64 + j * 16 + l15;
        v2f bv; bv.x = Xt[k0][nn]; bv.y = Xt[k0 + 1][nn];
#pragma unroll
        for (int i = 0; i < 2; ++i) acc[i][j] = wmma4(a[i], bv, acc[i][j]);
      }
    }
    __syncthreads();
  }
  float* yb = y2 + (size_t)b * COUT * HW;
#pragma unroll
  for (int i = 0; i < 2; ++i) {
    int mrow0 = m0 + wm * 32 + i * 16 + hi8;
#pragma unroll
    for (int j = 0; j < 4; ++j) {
      int nc = n0 + wn * 64 + j * 16 + l15;
#pragma unroll
      for (int r = 0; r < 8; ++r)
        yb[(size_t)(mrow0 + r) * HW + nc] = acc[i][j][r];
    }
  }
}

// ---------------------------------------------------------------------------
// Kernel G: out = gamma * relu(BN(y2)) + x
// ---------------------------------------------------------------------------
__global__ __launch_bounds__(256) void finalize(
    const float* __restrict__ y2, const float* __restrict__ scale,
    const float* __restrict__ shift, const float* __restrict__ x,
    const float* __restrict__ gamma, float* __restrict__ out) {
  size_t i = (size_t)blockIdx.x * 256 + threadIdx.x;
  if (i >= (size_t)BB * COUT * HW) return;
  int ch = (int)((i >> 12) & (COUT - 1));
  float v = y2[i] * scale[ch] + shift[ch];
  out[i] = gamma[0] * fmaxf(v, 0.f) + x[i];
}

// ---------------------------------------------------------------------------
extern "C" void kernel_launch(void* const* d_in, const int* in_sizes, int n_in,
                              void* d_out, int out_size, void* d_ws, size_t ws_size,
                              hipStream_t stream) {
  const float* x  = (const float*)d_in[0];
  const float* wf = (const float*)d_in[1];
  const float* gf = (const float*)d_in[2];
  const float* bf = (const float*)d_in[3];
  const float* wg = (const float*)d_in[4];
  const float* gg = (const float*)d_in[5];
  const float* bg = (const float*)d_in[6];
  const float* wh = (const float*)d_in[7];
  const float* gh = (const float*)d_in[8];
  const float* bh = (const float*)d_in[9];
  const float* wo = (const float*)d_in[10];
  const float* go = (const float*)d_in[11];
  const float* bo = (const float*)d_in[12];
  const float* gamma = (const float*)d_in[13];
  float* out = (float*)d_out;
  float* w = (float*)d_ws;

  // workspace layout (floats)
  size_t off = 0;
  float* y1     = w + off; off += (size_t)BB * CCAT * HW;   // 6,291,456
  float* wcat   = w + off; off += (size_t)CCAT * CIN;       //   196,608
  float* gcat   = w + off; off += CCAT;
  float* bcat   = w + off; off += CCAT;
  float* scale1 = w + off; off += CCAT;
  float* shift1 = w + off; off += CCAT;
  float* gact   = w + off; off += (size_t)BB * CQK * HW;    // 1,048,576
  float* fpool  = w + off; off += (size_t)BB * CQK * HWP;   //   262,144
  float* htb    = w + off; off += (size_t)BB * HWP * CH;    // 1,048,576
  float* oatt   = w + off; off += (size_t)BB * HW * CH;     // 4,194,304
  float* y2     = w + off; off += (size_t)BB * COUT * HW;   // 8,388,608
  float* scale2 = w + off; off += COUT;
  float* shift2 = w + off; off += COUT;                     // total ~86 MB

  pack_w<<<(CCAT * CIN + 255) / 256, 256, 0, stream>>>(
      wf, wg, wh, gf, bf, gg, bg, gh, bh, wcat, gcat, bcat);

  gemm_qkv<<<dim3(HW / 128, CCAT / 128, BB), 256, 0, stream>>>(x, wcat, y1);

  bn_stats<<<CCAT, 256, 0, stream>>>(y1, gcat, bcat, scale1, shift1, CCAT);

  {
    const int PER_B = CQK * HW + CQK * HWP + CH * HWP;
    act_pool<<<(BB * PER_B + 255) / 256, 256, 0, stream>>>(
        y1, scale1, shift1, gact, fpool, htb);
  }

  attention<<<dim3(HW / 64, BB), 256, 0, stream>>>(gact, fpool, htb, oatt);

  gemm_out<<<dim3(HW / 128, COUT / 128, BB), 256, 0, stream>>>(oatt, wo, y2);

  bn_stats<<<COUT, 256, 0, stream>>>(y2, go, bo, scale2, shift2, COUT);

  finalize<<<(int)(((size_t)BB * COUT * HW + 255) / 256), 256, 0, stream>>>(
      y2, scale2, shift2, x, gamma, out);
}